// TBasisWeightsFactory_40716289966667
// MI455X (gfx1250) — compile-verified
//
#include <hip/hip_runtime.h>
#include <math.h>

// CDNA5 (gfx1250) wave32 WMMA, f32 16x16x4.
typedef float v2f __attribute__((ext_vector_type(2)));
typedef float v8f __attribute__((ext_vector_type(8)));

__device__ __forceinline__ v8f wmma_f32_k4(v2f a, v2f b, v8f c) {
  // 8-arg form: (neg_a, A, neg_b, B, c_mod, C, reuse_a, reuse_b)
  return __builtin_amdgcn_wmma_f32_16x16x4_f32(false, a, false, b, (short)0, c,
                                               false, false);
}

// Problem constants (from the reference): NT=64, NM=8, S=256, R=16, M=6
// cores: [512][R=16][M=6][R=16] flat x = i*96 + m*16 + j  (1536 per core)
// level1: [p=4][b=64][i=16][mn=36][j=16]  (576 per (p,b,i))

// ---------------------------------------------------------------------------
// Kernel 1: cores = (W[512,256] @ TB[256,1536]) * exp(RA)[c, i]   (WMMA GEMM)
// one wave per 16x16 C tile; 32x96 tiles; K=256 as 64 chained 16x16x4 WMMAs.
// ---------------------------------------------------------------------------
__global__ __launch_bounds__(256) void k_assemble(const float* __restrict__ W,
                                                  const float* __restrict__ RA,
                                                  const float* __restrict__ TB,
                                                  float* __restrict__ cores) {
  const int lane = threadIdx.x & 31, wv = threadIdx.x >> 5;
  const int tile = blockIdx.x * 8 + wv;          // 384*8 = 3072 tiles
  const int mt = tile / 96, nt = tile % 96;
  const int h = lane >> 4, n16 = lane & 15;
  const int arow = (mt * 16 + n16) * 256;
  const int bcol = nt * 16 + n16;
  v8f acc = {};
#pragma unroll 4
  for (int kk = 0; kk < 256; kk += 4) {
    const int k0 = kk + 2 * h;
    v2f a = *(const v2f*)(W + arow + k0);                       // A[m][k0,k0+1]
    v2f b = {TB[k0 * 1536 + bcol], TB[(k0 + 1) * 1536 + bcol]}; // B[k][n]
    acc = wmma_f32_k4(a, b, acc);
  }
  // 96 = 6*16, so i-mode index is uniform within a 16-wide tile.
  const int imode = (nt * 16) / 96;
#pragma unroll
  for (int r = 0; r < 8; r++) {
    const int crow = mt * 16 + r + 8 * h;       // C: M = r + 8*(lane/16)
    cores[crow * 1536 + bcol] = acc[r] * __expf(RA[crow * 16 + imode]);
  }
}

// ---------------------------------------------------------------------------
// Kernel 2: round-1 pairwise contraction (8 -> 4 levels)
// L1[p][b][i][ma*6+mb][j] = sum_j1 cores[2p*64+b][i][ma][j1]*cores[(2p+1)*64+b][j1][mb][j]
// one wave per (p,b,mn): a single 16x16x16 matmul = 4 WMMAs.
// ---------------------------------------------------------------------------
__global__ __launch_bounds__(256) void k_round1(const float* __restrict__ cores,
                                                float* __restrict__ L1) {
  const int lane = threadIdx.x & 31, wv = threadIdx.x >> 5;
  const int tile = blockIdx.x * 8 + wv;          // 1152*8 = 9216 tiles
  const int p = tile / (64 * 36);
  const int rem = tile - p * 64 * 36;
  const int bb = rem / 36, mn = rem - (rem / 36) * 36;
  const int ma = mn / 6, mb = mn - ma * 6;
  const int h = lane >> 4, n16 = lane & 15;
  const float* A = cores + (size_t)(2 * p * 64 + bb) * 1536;
  const float* B = A + (size_t)64 * 1536;
  v8f acc = {};
#pragma unroll
  for (int ks = 0; ks < 4; ks++) {
    const int j1 = 4 * ks + 2 * h;
    v2f a = *(const v2f*)(A + n16 * 96 + ma * 16 + j1);   // rows i, K=j1
    v2f b = {B[j1 * 96 + mb * 16 + n16],                  // rows j1, cols j
             B[(j1 + 1) * 96 + mb * 16 + n16]};
    acc = wmma_f32_k4(a, b, acc);
  }
  float* o = L1 + (size_t)(p * 64 + bb) * (16 * 576);
#pragma unroll
  for (int r = 0; r < 8; r++)
    o[(r + 8 * h) * 576 + mn * 16 + n16] = acc[r];        // [i][mn][j]
}

// ---------------------------------------------------------------------------
// Kernel 3: fused round-2 + ring-trace GEMM.
// out[b, m*1296+n] = sum_{i,j} A2[b,i,m,j]*B2[b,j,n,i]
//   A2[b,:,m,:] = L1_0[b,:,m/36,:] @ L1_1[b,:,m%36,:]
//   B2[b,:,n,:] = L1_2[b,:,n/36,:] @ L1_3[b,:,n%36,:]
// Each 9-wave WG builds a 144x256 A' tile and 256x144 B' tile in LDS via
// WMMA, then runs the 144x144x256 tile GEMM (64 k-steps of 16x16x4).
// LDS: 2 * 144 * 258 * 4B = 297,216 B  (CDNA5: up to 320KB per workgroup)
// ---------------------------------------------------------------------------
#define TDIM 144
#define KPAD 258  // even (8B-aligned b64 frag loads), pitch!=mult of 64 banks

__global__ __launch_bounds__(288) void k_final(const float* __restrict__ L1,
                                               float* __restrict__ out) {
  __shared__ float At[TDIM * KPAD];  // A'[mLocal][i*16+j]
  __shared__ float Bt[TDIM * KPAD];  // B'[nLocal][i*16+j]  (n-major)
  const int lane = threadIdx.x & 31, wv = threadIdx.x >> 5;  // wv: 0..8
  const int h = lane >> 4, n16 = lane & 15;
  const int bb = blockIdx.z;
  const int m0 = blockIdx.y * TDIM, n0 = blockIdx.x * TDIM;
  const size_t PB = (size_t)64 * 16 * 576;
  const float* L10 = L1;
  const float* L11 = L1 + PB;
  const float* L12 = L1 + 2 * PB;
  const float* L13 = L1 + 3 * PB;

  // ---- build A' tile rows: wave wv handles mLocal = wv*16 .. wv*16+15 ----
  for (int t = 0; t < 16; t++) {
    const int mL = wv * 16 + t, mg = m0 + mL;
    const int mA = mg / 36, mB = mg - mA * 36;
    v8f acc = {};
#pragma unroll
    for (int ks = 0; ks < 4; ks++) {
      const int j1 = 4 * ks + 2 * h;
      v2f a = *(const v2f*)(L10 + ((size_t)bb * 16 + n16) * 576 + mA * 16 + j1);
      v2f b = {L11[((size_t)bb * 16 + j1) * 576 + mB * 16 + n16],
               L11[((size_t)bb * 16 + j1 + 1) * 576 + mB * 16 + n16]};
      acc = wmma_f32_k4(a, b, acc);
    }
    // result[i=r+8h][j=n16] -> At[mL][i*16+j]
#pragma unroll
    for (int r = 0; r < 8; r++)
      At[mL * KPAD + (r + 8 * h) * 16 + n16] = acc[r];
  }

  // ---- build B' tile cols (stored n-major, K index = i*16+j) ----
  for (int t = 0; t < 16; t++) {
    const int nL = wv * 16 + t, ng = n0 + nL;
    const int nA = ng / 36, nB = ng - nA * 36;
    v8f acc = {};
#pragma unroll
    for (int ks = 0; ks < 4; ks++) {
      const int j2 = 4 * ks + 2 * h;
      v2f a = *(const v2f*)(L12 + ((size_t)bb * 16 + n16) * 576 + nA * 16 + j2);
      v2f b = {L13[((size_t)bb * 16 + j2) * 576 + nB * 16 + n16],
               L13[((size_t)bb * 16 + j2 + 1) * 576 + nB * 16 + n16]};
      acc = wmma_f32_k4(a, b, acc);
    }
    // result[j=r+8h][i=n16] -> Bt[nL][i*16 + j]  (transpose into K order)
#pragma unroll
    for (int r = 0; r < 8; r++)
      Bt[nL * KPAD + n16 * 16 + r + 8 * h] = acc[r];
  }
  __syncthreads();

  // ---- main 144x144x256 tile GEMM: wave owns one 16-row strip x 9 C tiles
  v8f acc[9];
#pragma unroll
  for (int c = 0; c < 9; c++) acc[c] = (v8f){};
  const int arow = (wv * 16 + n16) * KPAD;
#pragma unroll 2
  for (int kk = 0; kk < 256; kk += 4) {
    const int k0 = kk + 2 * h;
    v2f a = *(const v2f*)(At + arow + k0);      // A frag reused for 9 WMMAs
#pragma unroll
    for (int c = 0; c < 9; c++) {
      v2f b = *(const v2f*)(Bt + (c * 16 + n16) * KPAD + k0);
      acc[c] = wmma_f32_k4(a, b, acc[c]);
    }
  }

  float* ob = out + (size_t)bb * 1296 * 1296;
#pragma unroll
  for (int c = 0; c < 9; c++) {
    const int ncol = n0 + c * 16 + n16;
#pragma unroll
    for (int r = 0; r < 8; r++) {
      const int mrow = m0 + wv * 16 + r + 8 * h;
      ob[(size_t)mrow * 1296 + ncol] = acc[c][r];
    }
  }
}

// ---------------------------------------------------------------------------
extern "C" void kernel_launch(void* const* d_in, const int* in_sizes, int n_in,
                              void* d_out, int out_size, void* d_ws,
                              size_t ws_size, hipStream_t stream) {
  (void)in_sizes; (void)n_in; (void)out_size; (void)ws_size;
  const float* W  = (const float*)d_in[0];   // [512,256]
  const float* RA = (const float*)d_in[1];   // [512,16]
  const float* TB = (const float*)d_in[2];   // [256,1536]
  float* out = (float*)d_out;                // [64, 6^8]
  float* cores = (float*)d_ws;               // 512*1536 f32 = 3.1 MB
  float* L1 = cores + (size_t)512 * 1536;    // 4*64*16*576 f32 = 9.4 MB

  k_assemble<<<384, 256, 0, stream>>>(W, RA, TB, cores);   // 3072 tiles / 8 waves
  k_round1<<<1152, 256, 0, stream>>>(cores, L1);           // 9216 tiles / 8 waves
  k_final<<<dim3(9, 9, 64), 288, 0, stream>>>(L1, out);    // 5184 WGs, 9 waves
}